// MessagePassingLayer_90744069030126
// MI455X (gfx1250) — compile-verified
//
#include <hip/hip_runtime.h>
#include <math.h>

// ---------------- problem constants (fixed by the reference) ----------------
#define DD 128
static constexpr int cN0 = 50000;
static constexpr int cE0 = 800000;
static constexpr int cK1 = 25000;   // ceil(0.5*50000)
static constexpr int cK2 = 12500;   // ceil(0.5*25000)
static constexpr float cEPS = 1e-12f;

typedef float v2f __attribute__((ext_vector_type(2)));
typedef float v8f __attribute__((ext_vector_type(8)));

// ---------------- generic fills ----------------
__global__ void k_fill_f(float* __restrict__ p, long n, float v) {
  long i = (long)blockIdx.x * blockDim.x + threadIdx.x;
  if (i < n) p[i] = v;
}
__global__ void k_fill_i(int* __restrict__ p, long n, int v) {
  long i = (long)blockIdx.x * blockDim.x + threadIdx.x;
  if (i < n) p[i] = v;
}
__global__ void k_fill_u(unsigned* __restrict__ p, long n, unsigned v) {
  long i = (long)blockIdx.x * blockDim.x + threadIdx.x;
  if (i < n) p[i] = v;
}

// ---------------- SAGE: scatter-mean over edges ----------------
// one thread per (edge, 4-float chunk): 32 chunks cover D=128
__global__ void k_scatter_mean(const float* __restrict__ x,
                               const int* __restrict__ src,
                               const int* __restrict__ dst,
                               const float* __restrict__ valid,
                               float* __restrict__ agg, float* __restrict__ cnt,
                               int nE) {
  long t = (long)blockIdx.x * blockDim.x + threadIdx.x;
  int e = (int)(t >> 5);
  if (e >= nE) return;
  int c = ((int)t & 31) * 4;
  float v = valid ? valid[e] : 1.0f;
  int s = src[e], d = dst[e];
  if (c == 0) atomicAdd(cnt + d, v);
  if (v == 0.0f) return;
  const float4 xv = *(const float4*)(x + (size_t)s * DD + c);
  float* a = agg + (size_t)d * DD + c;
  atomicAdd(a + 0, xv.x * v);
  atomicAdd(a + 1, xv.y * v);
  atomicAdd(a + 2, xv.z * v);
  atomicAdd(a + 3, xv.w * v);
}

__global__ void k_mean_div(float* __restrict__ agg, const float* __restrict__ cnt, int n) {
  long i = (long)blockIdx.x * blockDim.x + threadIdx.x;
  if (i >= (long)n * DD) return;
  int r = (int)(i >> 7);
  agg[i] /= fmaxf(cnt[r], 1.0f);
}

// ---------------- fused SAGE GEMM:  out = mean@Wl^T + bl + x@Wr^T ----------
// block = 256 thr = 8 waves; wave w -> columns [16w,16w+16); block -> rows [16*bx,...)
// V_WMMA_F32_16X16X4_F32, K stepped by 4 over D=128 (32 steps, 2 wmma each).
// Two independent accumulators (one per GEMM) break the WMMA RAW chain so the
// matrix pipe can overlap consecutive wmma ops; merged at the end.
__global__ __launch_bounds__(256) void k_sage_gemm(
    const float* __restrict__ mean, const float* __restrict__ xin,
    const float* __restrict__ Wl, const float* __restrict__ bl,
    const float* __restrict__ Wr, float* __restrict__ out, int n) {
  const int lane = threadIdx.x & 31;
  const int wv   = threadIdx.x >> 5;
  const int j0   = wv * 16;
  const int m0   = blockIdx.x * 16;
  const int lo16 = lane & 15;   // A: row m, B: col n, C: col n
  const int half = lane >> 4;   // 0/1 selects K pair (A/B) or row+8 (C)
  const int ka   = 2 * half;

  int rowA = m0 + lo16;
  size_t arow = (size_t)((rowA < n) ? rowA : 0) * DD;  // clamp; garbage rows not stored
  const float* wlr = Wl + (size_t)(j0 + lo16) * DD;
  const float* wrr = Wr + (size_t)(j0 + lo16) * DD;

  v8f accL, accR;
  const float b = bl[j0 + lo16];   // C column = j0 + lane%16 for all 8 row-VGPRs
#pragma unroll
  for (int v = 0; v < 8; ++v) { accL[v] = b; accR[v] = 0.0f; }

#pragma unroll 4
  for (int k = 0; k < DD; k += 4) {
    v2f a_m = *(const v2f*)(mean + arow + k + ka);
    v2f a_x = *(const v2f*)(xin  + arow + k + ka);
    v2f b_l = *(const v2f*)(wlr + k + ka);
    v2f b_r = *(const v2f*)(wrr + k + ka);
    accL = __builtin_amdgcn_wmma_f32_16x16x4_f32(false, a_m, false, b_l,
                                                 (short)0, accL, false, false);
    accR = __builtin_amdgcn_wmma_f32_16x16x4_f32(false, a_x, false, b_r,
                                                 (short)0, accR, false, false);
  }

#pragma unroll
  for (int v = 0; v < 8; ++v) accL[v] += accR[v];

  float* orow = out + (size_t)(m0 + 8 * half) * DD + j0 + lo16;
  if (m0 + 16 <= n) {          // uniform branch: full tile, unguarded strided stores
#pragma unroll
    for (int v = 0; v < 8; ++v) orow[(size_t)v * DD] = accL[v];
  } else {                     // partial last tile only
#pragma unroll
    for (int v = 0; v < 8; ++v) {
      int row = m0 + v + 8 * half;
      if (row < n) out[(size_t)row * DD + j0 + lo16] = accL[v];
    }
  }
}

// ---------------- pooling: score, radix-select, compaction ----------------
__global__ void k_pnorm(const float* __restrict__ p, float* __restrict__ rnorm) {
  __shared__ float s[DD];
  s[threadIdx.x] = p[threadIdx.x] * p[threadIdx.x];
  __syncthreads();
  for (int o = 64; o > 0; o >>= 1) {
    if (threadIdx.x < o) s[threadIdx.x] += s[threadIdx.x + o];
    __syncthreads();
  }
  if (threadIdx.x == 0) *rnorm = rsqrtf(fmaxf(s[0], 1e-30f));
}

__global__ void k_score(const float* __restrict__ x, const float* __restrict__ p,
                        const float* __restrict__ rnorm, float* __restrict__ score,
                        unsigned* __restrict__ keys, int n) {
  int i = blockIdx.x * blockDim.x + threadIdx.x;
  if (i >= n) return;
  const float* xr = x + (size_t)i * DD;
  float s = 0.f;
#pragma unroll 8
  for (int k = 0; k < DD; ++k) s += xr[k] * p[k];
  s *= *rnorm;
  score[i] = s;
  unsigned u = __float_as_uint(s);                 // orderable-bits transform
  keys[i] = (u & 0x80000000u) ? ~u : (u | 0x80000000u);
}

__global__ void k_state_init(unsigned* st, unsigned k) { st[0] = 0u; st[1] = 0u; st[2] = k; }

__global__ void k_hist(const unsigned* __restrict__ keys, int n,
                       const unsigned* __restrict__ st, int shift,
                       unsigned* __restrict__ hist) {
  int i = blockIdx.x * blockDim.x + threadIdx.x;
  if (i >= n) return;
  unsigned key = keys[i];
  if ((key & st[1]) == st[0]) atomicAdd(&hist[(key >> shift) & 0xFFu], 1u);
}

__global__ void k_radix_select(unsigned* st, const unsigned* __restrict__ hist, int shift) {
  unsigned krem = st[2], b = 0u;
  for (int i = 255; i >= 0; --i) {
    unsigned h = hist[i];
    if (krem <= h) { b = (unsigned)i; break; }
    krem -= h;
  }
  st[0] |= (b << shift);
  st[1] |= (0xFFu << shift);
  st[2] = krem;
}

__global__ void k_compact_gt(const unsigned* __restrict__ keys, int n,
                             const unsigned* __restrict__ st, int* __restrict__ perm,
                             unsigned* ctrGT) {
  int i = blockIdx.x * blockDim.x + threadIdx.x;
  if (i >= n) return;
  if (keys[i] > st[0]) perm[atomicAdd(ctrGT, 1u)] = i;
}
__global__ void k_compact_eq(const unsigned* __restrict__ keys, int n,
                             const unsigned* __restrict__ st, int* __restrict__ perm,
                             const unsigned* __restrict__ ctrGT, unsigned* ctrEQ, int k) {
  int i = blockIdx.x * blockDim.x + threadIdx.x;
  if (i >= n) return;
  if (keys[i] == st[0]) {
    unsigned p = atomicAdd(ctrEQ, 1u);
    unsigned g = *ctrGT;                 // final after previous kernel (stream order)
    if (g + p < (unsigned)k) perm[g + p] = i;
  }
}

__global__ void k_set_newid(const int* __restrict__ perm, int k, int* __restrict__ newid) {
  int j = blockIdx.x * blockDim.x + threadIdx.x;
  if (j < k) newid[perm[j]] = j;
}

__global__ void k_pool_gather(const float* __restrict__ xin, const float* __restrict__ score,
                              const int* __restrict__ perm, float* __restrict__ xout, int k) {
  long t = (long)blockIdx.x * blockDim.x + threadIdx.x;
  int j = (int)(t >> 5);
  if (j >= k) return;
  int c = ((int)t & 31) * 4;
  int p = perm[j];
  float g = tanhf(score[p]);
  float4 v = *(const float4*)(xin + (size_t)p * DD + c);
  float4 o = make_float4(v.x * g, v.y * g, v.z * g, v.w * g);
  *(float4*)(xout + (size_t)j * DD + c) = o;
}

__global__ void k_gather_w(const float* __restrict__ win, const int* __restrict__ perm,
                           float* __restrict__ wout, int k) {
  int j = blockIdx.x * blockDim.x + threadIdx.x;
  if (j < k) wout[j] = win[perm[j]];
}

__global__ void k_remap(const int* __restrict__ src, const int* __restrict__ dst,
                        const float* __restrict__ valid, const int* __restrict__ newid,
                        int* __restrict__ src2, int* __restrict__ dst2,
                        float* __restrict__ valid2, int nE) {
  int e = blockIdx.x * blockDim.x + threadIdx.x;
  if (e >= nE) return;
  int s = newid[src[e]], d = newid[dst[e]];
  float v = valid ? valid[e] : 1.0f;
  float keep = (s >= 0 && d >= 0) ? 1.0f : 0.0f;
  valid2[e] = v * keep;
  src2[e] = s < 0 ? 0 : s;
  dst2[e] = d < 0 ? 0 : d;
}

// ---------------- weighted edge conv ----------------
__global__ void k_deg(const int* __restrict__ src, const float* __restrict__ valid,
                      float* __restrict__ deg, int nE) {
  int e = blockIdx.x * blockDim.x + threadIdx.x;
  if (e >= nE) return;
  float v = valid ? valid[e] : 1.0f;
  if (v != 0.0f) atomicAdd(&deg[src[e]], v);
}
__global__ void k_wsend(const int* __restrict__ src, const int* __restrict__ dst,
                        const float* __restrict__ valid, const float* __restrict__ w,
                        const float* __restrict__ deg, float* __restrict__ ectmp,
                        float* __restrict__ wsum, int nE) {
  int e = blockIdx.x * blockDim.x + threadIdx.x;
  if (e >= nE) return;
  float v = valid ? valid[e] : 1.0f;
  int s = src[e];
  float dg = deg[s];
  float t = w[s] / (dg > 0.0f ? dg : 1.0f) * v;
  ectmp[e] = t;
  if (t != 0.0f) atomicAdd(&wsum[dst[e]], t);
}
__global__ void k_add_eps(float* __restrict__ wsum, int n) {
  int i = blockIdx.x * blockDim.x + threadIdx.x;
  if (i < n) wsum[i] += cEPS;
}
__global__ void k_ec(const int* __restrict__ dst, const float* __restrict__ ectmp,
                     const float* __restrict__ wsum, float* __restrict__ ec, int nE) {
  int e = blockIdx.x * blockDim.x + threadIdx.x;
  if (e >= nE) return;
  ec[e] = ectmp[e] / wsum[dst[e]];
}

// gather at index a[e], scatter-add at index b[e] (down: a=src,b=dst; up: a=dst,b=src)
__global__ void k_econv(const float* __restrict__ x, const int* __restrict__ a,
                        const int* __restrict__ b, const float* __restrict__ ec,
                        float* __restrict__ out, int nE) {
  long t = (long)blockIdx.x * blockDim.x + threadIdx.x;
  int e = (int)(t >> 5);
  if (e >= nE) return;
  int c = ((int)t & 31) * 4;
  float w = ec[e];
  if (w == 0.0f) return;
  const float4 v = *(const float4*)(x + (size_t)a[e] * DD + c);
  float* o = out + (size_t)b[e] * DD + c;
  atomicAdd(o + 0, v.x * w);
  atomicAdd(o + 1, v.y * w);
  atomicAdd(o + 2, v.z * w);
  atomicAdd(o + 3, v.w * w);
}

__global__ void k_unpool(const float* __restrict__ xin, const int* __restrict__ perm,
                         float* __restrict__ xout, int k) {
  long t = (long)blockIdx.x * blockDim.x + threadIdx.x;
  int j = (int)(t >> 5);
  if (j >= k) return;
  int c = ((int)t & 31) * 4;
  *(float4*)(xout + (size_t)perm[j] * DD + c) = *(const float4*)(xin + (size_t)j * DD + c);
}

__global__ void k_add_inplace(float* __restrict__ x, const float* __restrict__ y, long n) {
  long i = (long)blockIdx.x * blockDim.x + threadIdx.x;
  if (i < n) x[i] += y[i];
}
__global__ void k_add_out(const float* __restrict__ a, const float* __restrict__ b,
                          float* __restrict__ o, long n) {
  long i = (long)blockIdx.x * blockDim.x + threadIdx.x;
  if (i < n) o[i] = a[i] + b[i];
}

// =============================== host ======================================
extern "C" void kernel_launch(void* const* d_in, const int* in_sizes, int n_in,
                              void* d_out, int out_size, void* d_ws, size_t ws_size,
                              hipStream_t stream) {
  (void)in_sizes; (void)n_in; (void)out_size; (void)ws_size;
  const float* x0  = (const float*)d_in[0];
  const int*   ei  = (const int*)d_in[1];
  const float* Wl  = (const float*)d_in[2];
  const float* bl  = (const float*)d_in[3];
  const float* Wr  = (const float*)d_in[4];
  const float* pp  = (const float*)d_in[5];
  float* out = (float*)d_out;
  const int* src0 = ei;
  const int* dst0 = ei + cE0;

  char* ws = (char*)d_ws; size_t off = 0;
  auto alloc = [&](size_t b) -> char* {
    char* p = ws + off; off = (off + b + 255) & ~(size_t)255; return p;
  };
  float*    bufX  = (float*)alloc((size_t)cN0 * DD * 4);
  float*    bufY  = (float*)alloc((size_t)cN0 * DD * 4);
  float*    agg   = (float*)alloc((size_t)cN0 * DD * 4);
  float*    down0 = (float*)alloc((size_t)cN0 * DD * 4);
  float*    down1 = (float*)alloc((size_t)cK1 * DD * 4);
  float*    cnt   = (float*)alloc((size_t)cN0 * 4);
  float*    deg   = (float*)alloc((size_t)cN0 * 4);
  float*    wsum  = (float*)alloc((size_t)cN0 * 4);
  float*    w0    = (float*)alloc((size_t)cN0 * 4);
  float*    w1    = (float*)alloc((size_t)cN0 * 4);
  float*    score = (float*)alloc((size_t)cN0 * 4);
  unsigned* keys  = (unsigned*)alloc((size_t)cN0 * 4);
  int*      newid = (int*)alloc((size_t)cN0 * 4);
  int*      perm1 = (int*)alloc((size_t)cK1 * 4);
  int*      perm2 = (int*)alloc((size_t)cK2 * 4);
  int*      src1  = (int*)alloc((size_t)cE0 * 4);
  int*      dst1  = (int*)alloc((size_t)cE0 * 4);
  float*    vld1  = (float*)alloc((size_t)cE0 * 4);
  int*      src2  = (int*)alloc((size_t)cE0 * 4);
  int*      dst2  = (int*)alloc((size_t)cE0 * 4);
  float*    vld2  = (float*)alloc((size_t)cE0 * 4);
  float*    ec0   = (float*)alloc((size_t)cE0 * 4);
  float*    ec1   = (float*)alloc((size_t)cE0 * 4);
  float*    ectmp = (float*)alloc((size_t)cE0 * 4);
  unsigned* hist  = (unsigned*)alloc(256 * 4);
  unsigned* state = (unsigned*)alloc(64 * 4);  // [0]prefix [1]mask [2]krem [3]ctrGT [4]ctrEQ
  float*    rnorm = (float*)alloc(64 * 4);

  auto gb = [](long n, int b) { return dim3((unsigned)((n + b - 1) / b)); };

  auto sage = [&](const float* xin, float* xout, const int* s, const int* d,
                  const float* valid, int n, int conv) {
    k_fill_f<<<gb((long)n * DD, 256), 256, 0, stream>>>(agg, (long)n * DD, 0.f);
    k_fill_f<<<gb(n, 256), 256, 0, stream>>>(cnt, n, 0.f);
    k_scatter_mean<<<gb((long)cE0 * 32, 256), 256, 0, stream>>>(xin, s, d, valid, agg, cnt, cE0);
    k_mean_div<<<gb((long)n * DD, 256), 256, 0, stream>>>(agg, cnt, n);
    k_sage_gemm<<<dim3((unsigned)((n + 15) / 16)), 256, 0, stream>>>(
        agg, xin, Wl + (size_t)conv * DD * DD, bl + (size_t)conv * DD,
        Wr + (size_t)conv * DD * DD, xout, n);
  };

  auto cal_ew = [&](const int* s, const int* d, const float* valid, const float* w,
                    int n, float* ec) {  // leaves new w (= aggr_w) in wsum
    k_fill_f<<<gb(n, 256), 256, 0, stream>>>(deg, n, 0.f);
    k_deg<<<gb(cE0, 256), 256, 0, stream>>>(s, valid, deg, cE0);
    k_fill_f<<<gb(n, 256), 256, 0, stream>>>(wsum, n, 0.f);
    k_wsend<<<gb(cE0, 256), 256, 0, stream>>>(s, d, valid, w, deg, ectmp, wsum, cE0);
    k_add_eps<<<gb(n, 256), 256, 0, stream>>>(wsum, n);
    k_ec<<<gb(cE0, 256), 256, 0, stream>>>(d, ectmp, wsum, ec, cE0);
  };

  auto pool = [&](const float* xin, float* xout, const float* p, int n, int k,
                  int* perm, const float* win, float* wout,
                  const int* sIn, const int* dIn, const float* vIn,
                  int* sOut, int* dOut, float* vOut) {
    k_pnorm<<<1, 128, 0, stream>>>(p, rnorm);
    k_score<<<gb(n, 256), 256, 0, stream>>>(xin, p, rnorm, score, keys, n);
    k_state_init<<<1, 1, 0, stream>>>(state, (unsigned)k);
    for (int shift = 24; shift >= 0; shift -= 8) {
      k_fill_u<<<1, 256, 0, stream>>>(hist, 256, 0u);
      k_hist<<<gb(n, 256), 256, 0, stream>>>(keys, n, state, shift, hist);
      k_radix_select<<<1, 1, 0, stream>>>(state, hist, shift);
    }
    k_fill_u<<<1, 64, 0, stream>>>(state + 3, 2, 0u);
    k_compact_gt<<<gb(n, 256), 256, 0, stream>>>(keys, n, state, perm, state + 3);
    k_compact_eq<<<gb(n, 256), 256, 0, stream>>>(keys, n, state, perm, state + 3, state + 4, k);
    k_fill_i<<<gb(n, 256), 256, 0, stream>>>(newid, n, -1);
    k_set_newid<<<gb(k, 256), 256, 0, stream>>>(perm, k, newid);
    k_pool_gather<<<gb((long)k * 32, 256), 256, 0, stream>>>(xin, score, perm, xout, k);
    if (wout) k_gather_w<<<gb(k, 256), 256, 0, stream>>>(win, perm, wout, k);
    k_remap<<<gb(cE0, 256), 256, 0, stream>>>(sIn, dIn, vIn, newid, sOut, dOut, vOut, cE0);
  };

  // ---- init ----
  k_fill_f<<<gb(cN0, 256), 256, 0, stream>>>(w0, cN0, 1.f);

  // ---- down level 0 ----
  sage(x0,   bufX, src0, dst0, nullptr, cN0, 0);
  sage(bufX, bufY, src0, dst0, nullptr, cN0, 1);
  hipMemcpyAsync(down0, bufY, (size_t)cN0 * DD * 4, hipMemcpyDeviceToDevice, stream);
  cal_ew(src0, dst0, nullptr, w0, cN0, ec0);
  k_fill_f<<<gb((long)cN0 * DD, 256), 256, 0, stream>>>(bufX, (long)cN0 * DD, 0.f);
  k_econv<<<gb((long)cE0 * 32, 256), 256, 0, stream>>>(bufY, src0, dst0, ec0, bufX, cE0);
  pool(bufX, bufY, pp, cN0, cK1, perm1, wsum, w1, src0, dst0, nullptr, src1, dst1, vld1);

  // ---- down level 1 ----
  sage(bufY, bufX, src1, dst1, vld1, cK1, 2);
  sage(bufX, bufY, src1, dst1, vld1, cK1, 3);
  hipMemcpyAsync(down1, bufY, (size_t)cK1 * DD * 4, hipMemcpyDeviceToDevice, stream);
  cal_ew(src1, dst1, vld1, w1, cK1, ec1);
  k_fill_f<<<gb((long)cK1 * DD, 256), 256, 0, stream>>>(bufX, (long)cK1 * DD, 0.f);
  k_econv<<<gb((long)cE0 * 32, 256), 256, 0, stream>>>(bufY, src1, dst1, ec1, bufX, cE0);
  pool(bufX, bufY, pp + DD, cK1, cK2, perm2, wsum, nullptr, src1, dst1, vld1, src2, dst2, vld2);

  // ---- bottom ----
  sage(bufY, bufX, src2, dst2, vld2, cK2, 4);
  sage(bufX, bufY, src2, dst2, vld2, cK2, 5);

  // ---- up u=1 ----
  k_fill_f<<<gb((long)cK1 * DD, 256), 256, 0, stream>>>(bufX, (long)cK1 * DD, 0.f);
  k_unpool<<<gb((long)cK2 * 32, 256), 256, 0, stream>>>(bufY, perm2, bufX, cK2);
  k_fill_f<<<gb((long)cK1 * DD, 256), 256, 0, stream>>>(bufY, (long)cK1 * DD, 0.f);
  k_econv<<<gb((long)cE0 * 32, 256), 256, 0, stream>>>(bufX, dst1, src1, ec1, bufY, cE0);
  sage(bufY, bufX, src1, dst1, vld1, cK1, 6);
  sage(bufX, bufY, src1, dst1, vld1, cK1, 7);
  k_add_inplace<<<gb((long)cK1 * DD, 256), 256, 0, stream>>>(bufY, down1, (long)cK1 * DD);

  // ---- up u=0 ----
  k_fill_f<<<gb((long)cN0 * DD, 256), 256, 0, stream>>>(bufX, (long)cN0 * DD, 0.f);
  k_unpool<<<gb((long)cK1 * 32, 256), 256, 0, stream>>>(bufY, perm1, bufX, cK1);
  k_fill_f<<<gb((long)cN0 * DD, 256), 256, 0, stream>>>(bufY, (long)cN0 * DD, 0.f);
  k_econv<<<gb((long)cE0 * 32, 256), 256, 0, stream>>>(bufX, dst0, src0, ec0, bufY, cE0);
  sage(bufY, bufX, src0, dst0, nullptr, cN0, 8);
  sage(bufX, bufY, src0, dst0, nullptr, cN0, 9);
  k_add_out<<<gb((long)cN0 * DD, 256), 256, 0, stream>>>(bufY, down0, out, (long)cN0 * DD);
}